// Sampler_42545946034922
// MI455X (gfx1250) — compile-verified
//
#include <hip/hip_runtime.h>

// ---------------------------------------------------------------------------
// Batched epsilon-greedy multinomial sampler, MI455X (gfx1250, wave32).
//
// argmax(log(w/S)+g) == argmax(log w + g) == argmin(-log(u)/w)  (exp race)
// -> no row-sum, one v_log_f32 + one v_rcp_f32 (both TRANS, co-execute with
// VALU) per element. JAX threefry2x32-20 bitstream reproduced; counter
// pairing (i, i+2^26) means rows r and r+2^19 share threefry calls, so each
// wave processes that row pair. The per-pair epsilon-coin threefry runs on
// the SALU (scalar pair index via readfirstlane) and co-issues for free.
//
// CDNA5 paths: GLOBAL_LOAD_ASYNC_TO_LDS_B128 (saddr form, ASYNCcnt, double
// buffered), s_wait_asynccnt, ds_load_b128 tile reads, wave32 shuffle-tree
// argmin on packed u64 (valueBits<<32 | classIdx) keys.
// ---------------------------------------------------------------------------

#define SAMPLER_EPS 0.2f

static constexpr unsigned N_ROWS = 1u << 20;    // 1048576
static constexpr unsigned HALF   = N_ROWS >> 1; // 524288 row pairs
static constexpr unsigned NBLK   = 8192;        // 8 waves/block -> 65536 waves
static constexpr float    F32_TINY = 1.1754944e-38f;

struct TF2 { unsigned x0, x1; };

__host__ __device__ constexpr unsigned rotl32c(unsigned x, int r) {
  return (x << r) | (x >> (32 - r));
}

// Threefry-2x32, 20 rounds (JAX default PRNG).
__host__ __device__ constexpr TF2 threefry2x32(unsigned k0, unsigned k1,
                                               unsigned c0, unsigned c1) {
  const unsigned ks2 = k0 ^ k1 ^ 0x1BD11BDAu;
  unsigned x0 = c0 + k0;
  unsigned x1 = c1 + k1;
  const int r0[4] = {13, 15, 26, 6};
  const int r1[4] = {17, 29, 16, 24};
  for (int i = 0; i < 4; ++i) { x0 += x1; x1 = rotl32c(x1, r0[i]); x1 ^= x0; }
  x0 += k1;  x1 += ks2 + 1u;
  for (int i = 0; i < 4; ++i) { x0 += x1; x1 = rotl32c(x1, r1[i]); x1 ^= x0; }
  x0 += ks2; x1 += k0 + 2u;
  for (int i = 0; i < 4; ++i) { x0 += x1; x1 = rotl32c(x1, r0[i]); x1 ^= x0; }
  x0 += k0;  x1 += k1 + 3u;
  for (int i = 0; i < 4; ++i) { x0 += x1; x1 = rotl32c(x1, r1[i]); x1 ^= x0; }
  x0 += k1;  x1 += ks2 + 4u;
  for (int i = 0; i < 4; ++i) { x0 += x1; x1 = rotl32c(x1, r0[i]); x1 ^= x0; }
  x0 += ks2; x1 += k0 + 5u;
  return {x0, x1};
}

// split(key(42)) at compile time: bits = threefry((0,42), iota(4)),
// pairs (0,2) and (1,3); k_u = (a0,a1), k_g = (b0,b1).
static constexpr TF2 SPL0 = threefry2x32(0u, 42u, 0u, 2u);
static constexpr TF2 SPL1 = threefry2x32(0u, 42u, 1u, 3u);
static constexpr unsigned KU0 = SPL0.x0, KU1 = SPL1.x0;   // k_u
static constexpr unsigned KG0 = SPL0.x1, KG1 = SPL1.x1;   // k_g

__device__ __forceinline__ float bits_to_unit(unsigned b) {
  // JAX float32 path: bitcast(b>>9 | 0x3f800000) - 1  ->  [0,1)
  return __uint_as_float((b >> 9) | 0x3f800000u) - 1.0f;
}

__global__ __launch_bounds__(256) void
Sampler_42545946034922_kernel(const float* __restrict__ pmfs,
                              long long* __restrict__ out) {
  // [wave][buffer][row-of-pair][class] : 16 KB per block
  __shared__ float buf[8][2][2][128];

  const unsigned lane = threadIdx.x & 31u;
  const unsigned wv   = threadIdx.x >> 5;
  // per-wave-uniform -> force scalar so the pair loop, coin threefry and
  // prefetch base addresses all live on the SALU / SGPRs.
  const unsigned gw = __builtin_amdgcn_readfirstlane(blockIdx.x * 8u + wv);
  const unsigned NW = NBLK * 8u;

  // LDS byte offsets (workgroup-relative) for this lane's 16B slot in each buffer.
  unsigned ldsA[2], ldsB[2];
#pragma unroll
  for (int b = 0; b < 2; ++b) {
    ldsA[b] = (unsigned)(unsigned long long)(const void*)&buf[wv][b][0][lane * 4u];
    ldsB[b] = (unsigned)(unsigned long long)(const void*)&buf[wv][b][1][lane * 4u];
  }
  const unsigned voff = lane * 16u;

  auto prefetch = [&](int b, unsigned pair) {
    const char* baseA = (const char*)pmfs + ((unsigned long long)pair << 9);
    const char* baseB = (const char*)pmfs + ((unsigned long long)(pair + HALF) << 9);
    asm volatile("global_load_async_to_lds_b128 %0, %1, %2"
                 :: "v"(ldsA[b]), "v"(voff), "s"(baseA) : "memory");
    asm volatile("global_load_async_to_lds_b128 %0, %1, %2"
                 :: "v"(ldsB[b]), "v"(voff), "s"(baseB) : "memory");
  };

  int cur = 0;
  prefetch(0, gw);  // gw < 65536 <= HALF always

  for (unsigned p = gw; p < HALF; p += NW) {
    const unsigned pn = p + NW;
    if (pn < HALF) {
      prefetch(cur ^ 1, pn);                             // 2 more in flight
      asm volatile("s_wait_asynccnt 0x2" ::: "memory");  // current tile done
    } else {
      asm volatile("s_wait_asynccnt 0x0" ::: "memory");
    }

    const float4 wA4 = *reinterpret_cast<const float4*>(&buf[wv][cur][0][lane * 4u]);
    const float4 wB4 = *reinterpret_cast<const float4*>(&buf[wv][cur][1][lane * 4u]);
    const float wAv[4] = {wA4.x, wA4.y, wA4.z, wA4.w};
    const float wBv[4] = {wB4.x, wB4.y, wB4.z, wB4.w};

    // epsilon coins: scalar inputs -> SALU threefry, co-issues with VALU.
    const TF2 ub = threefry2x32(KU0, KU1, p, p + HALF);
    const bool epsA = bits_to_unit(ub.x0) < SAMPLER_EPS;
    const bool epsB = bits_to_unit(ub.x1) < SAMPLER_EPS;

    // packed race keys: high word = positive-float bits (order-preserving),
    // low word = class index (smallest index wins ties for free).
    unsigned long long kA = ~0ull, kB = ~0ull;

#pragma unroll
    for (int j = 0; j < 4; ++j) {
      const unsigned c   = lane * 4u + (unsigned)j;
      const unsigned idx = p * 128u + c;            // < 2^26
      // gumbel bits: counter pair (idx, idx + 2^26) == rows p, p+HALF
      const TF2 g = threefry2x32(KG0, KG1, idx, idx + (1u << 26));

      const float uA = fmaxf(bits_to_unit(g.x0), F32_TINY);
      const float uB = fmaxf(bits_to_unit(g.x1), F32_TINY);
      const float eA = -__logf(uA);                 // Exp(1)  (TRANS)
      const float eB = -__logf(uB);
      // e / w via raw v_rcp_f32 (TRANS, co-executes; order-only use)
      const float tA = epsA ? eA : eA * __builtin_amdgcn_rcpf(wAv[j]);
      const float tB = epsB ? eB : eB * __builtin_amdgcn_rcpf(wBv[j]);

      const unsigned long long pkA =
          ((unsigned long long)__float_as_uint(tA) << 32) | c;
      const unsigned long long pkB =
          ((unsigned long long)__float_as_uint(tB) << 32) | c;
      kA = pkA < kA ? pkA : kA;
      kB = pkB < kB ? pkB : kB;
    }

    // wave32 shuffle-tree min over packed keys
#pragma unroll
    for (int off = 16; off > 0; off >>= 1) {
      const unsigned long long oA =
          (unsigned long long)__shfl_xor((long long)kA, off, 32);
      const unsigned long long oB =
          (unsigned long long)__shfl_xor((long long)kB, off, 32);
      kA = oA < kA ? oA : kA;
      kB = oB < kB ? oB : kB;
    }

    if (lane == 0) {
      out[p]        = (long long)(unsigned)(kA & 0xffffffffu);
      out[p + HALF] = (long long)(unsigned)(kB & 0xffffffffu);
    }
    cur ^= 1;
  }
}

extern "C" void kernel_launch(void* const* d_in, const int* in_sizes, int n_in,
                              void* d_out, int out_size, void* d_ws, size_t ws_size,
                              hipStream_t stream) {
  (void)in_sizes; (void)n_in; (void)d_ws; (void)ws_size; (void)out_size;
  const float* pmfs = (const float*)d_in[0];
  long long*   out  = (long long*)d_out;
  Sampler_42545946034922_kernel<<<NBLK, 256, 0, stream>>>(pmfs, out);
}